// HardBatchMiningTripletLoss_16406775070902
// MI455X (gfx1250) — compile-verified
//
#include <hip/hip_runtime.h>
#include <hip/hip_bf16.h>

// CDNA5 wave32 WMMA fragment types: f32 16x16x4 -> A/B = 2 VGPRs, C/D = 8 VGPRs
typedef __attribute__((ext_vector_type(2))) float v2f;
typedef __attribute__((ext_vector_type(8))) float v8f;

#define BATCH   8192
#define DIM     256
#define MARGINF 0.3f
#define EPSF    1e-12f

#define BLK_M    128    // rows per block: 8 waves x 16
#define TN       32     // column tile per iteration (2 WMMA N-tiles)
#define JCHUNK   512    // columns per block (grid.y = BATCH/JCHUNK = 16)
#define LDSTRIDE 260    // 256 + 4 pad floats -> conflict-free b64 fragment reads
#define KSTEPS   (DIM / 4)  // 64 WMMA K-steps

#define USE_ASYNC_LDS 1  // GLOBAL_LOAD_ASYNC_TO_LDS_B128 staging (no VGPR staging)

// ---------------------------------------------------------------- init
__global__ __launch_bounds__(256) void tl_init(int* __restrict__ posA,
                                               int* __restrict__ negA,
                                               float* __restrict__ out) {
  int i = blockIdx.x * blockDim.x + threadIdx.x;
  if (i < BATCH) {
    posA[i] = 0xFF800000;  // -inf bits (signed-int max works: all d2 > 0)
    negA[i] = 0x7F800000;  // +inf bits
  }
  if (i == 0) out[0] = 0.0f;
}

// ---------------------------------------------------------------- normalize
__global__ __launch_bounds__(256) void tl_normalize(const float* __restrict__ x,
                                                    float* __restrict__ vs,
                                                    float* __restrict__ sq) {
  int row  = blockIdx.x * 8 + (threadIdx.x >> 5);
  int lane = threadIdx.x & 31;
  const float4* xr = reinterpret_cast<const float4*>(x + (size_t)row * DIM) + lane * 2;
  float4 a = xr[0], b = xr[1];
  float s = a.x * a.x + a.y * a.y + a.z * a.z + a.w * a.w +
            b.x * b.x + b.y * b.y + b.z * b.z + b.w * b.w;
#pragma unroll
  for (int m = 16; m >= 1; m >>= 1) s += __shfl_xor(s, m, 32);
  float inv = 1.0f / sqrtf(s);
  a.x *= inv; a.y *= inv; a.z *= inv; a.w *= inv;
  b.x *= inv; b.y *= inv; b.z *= inv; b.w *= inv;
  float s2 = a.x * a.x + a.y * a.y + a.z * a.z + a.w * a.w +
             b.x * b.x + b.y * b.y + b.z * b.z + b.w * b.w;
#pragma unroll
  for (int m = 16; m >= 1; m >>= 1) s2 += __shfl_xor(s2, m, 32);
  float4* o = reinterpret_cast<float4*>(vs + (size_t)row * DIM) + lane * 2;
  o[0] = a; o[1] = b;
  if (lane == 0) sq[row] = s2;
}

// ---------------------------------------------------------------- fused gram + hard mining
// Wave owns a 16-row M strip whose A fragments live in 128 VGPRs for the whole
// j-loop (no global loads in the K-loop). Per 32-column tile: B staged to LDS
// via GLOBAL_LOAD_ASYNC_TO_LDS_B128, then 64 K-steps x 2 independent WMMAs.
__global__ __launch_bounds__(256) void tl_mine(const float* __restrict__ vs,
                                               const float* __restrict__ sq,
                                               const int* __restrict__ tgt,
                                               int* __restrict__ posA,
                                               int* __restrict__ negA) {
  __shared__ float Bs[TN * LDSTRIDE];  // 33.3 KB

  const int lane  = threadIdx.x & 31;
  const int wave  = threadIdx.x >> 5;
  const int n     = lane & 15;   // WMMA column index within a 16-tile
  const int khalf = lane >> 4;   // which K pair within the 4-wide K step
  const int i0    = blockIdx.x * BLK_M + wave * 16;
  const int jbase = blockIdx.y * JCHUNK;

  // Rows this lane accumulates in the C layout: i = i0 + v + 8*khalf, v=0..7
  float sqi[8]; int ti[8];
#pragma unroll
  for (int v = 0; v < 8; ++v) {
    int r  = i0 + v + 8 * khalf;
    sqi[v] = sq[r];
    ti[v]  = tgt[r];
  }
  float pos[8], neg[8];
#pragma unroll
  for (int v = 0; v < 8; ++v) { pos[v] = -INFINITY; neg[v] = INFINITY; }

  // ---- Load the wave's full A strip into registers (64 x v2f = 128 VGPRs).
  const v2f* ap = reinterpret_cast<const v2f*>(vs + (size_t)(i0 + n) * DIM) + khalf;
  v2f areg[KSTEPS];
#pragma unroll
  for (int kk = 0; kk < KSTEPS; ++kk) areg[kk] = ap[2 * kk];

  // B fragment bases in LDS: column n (tile 0) and n+16 (tile 1)
  const v2f* bp0 = reinterpret_cast<const v2f*>(Bs) + n * (LDSTRIDE / 2) + khalf;
  const v2f* bp1 = bp0 + 16 * (LDSTRIDE / 2);

  // Staging assignment: thread -> (column c, 32-float k chunk)
  const int sc = threadIdx.x & 31;
  const int so = (threadIdx.x >> 5) * 32;
#if USE_ASYNC_LDS
  const unsigned lds_base =
      (unsigned)(uintptr_t)(Bs + sc * LDSTRIDE + so);  // low 32 bits = LDS offset
#endif

  for (int j0 = jbase; j0 < jbase + JCHUNK; j0 += TN) {
    const float* gbase = vs + (size_t)(j0 + sc) * DIM + so;
#if USE_ASYNC_LDS
    __syncthreads();  // previous tile's fragment reads complete before overwrite
    // 8 x b128 async copies: same INST_OFFSET applies to global and LDS address.
#pragma unroll
    for (int e = 0; e < 8; ++e)
      asm volatile("global_load_async_to_lds_b128 %0, %1, off offset:%2"
                   :: "v"(lds_base), "v"(gbase), "n"(e * 16) : "memory");
    asm volatile("s_wait_asynccnt 0x0" ::: "memory");
    __syncthreads();  // all waves' tiles visible
#else
    const float4* gsrc = reinterpret_cast<const float4*>(gbase);
    float4 q0 = gsrc[0], q1 = gsrc[1], q2 = gsrc[2], q3 = gsrc[3];
    __syncthreads();
    float2* dst = reinterpret_cast<float2*>(Bs + sc * LDSTRIDE + so);
    dst[0] = make_float2(q0.x, q0.y); dst[1] = make_float2(q0.z, q0.w);
    dst[2] = make_float2(q1.x, q1.y); dst[3] = make_float2(q1.z, q1.w);
    dst[4] = make_float2(q2.x, q2.y); dst[5] = make_float2(q2.z, q2.w);
    dst[6] = make_float2(q3.x, q3.y); dst[7] = make_float2(q3.z, q3.w);
    gsrc += 4;
    q0 = gsrc[0]; q1 = gsrc[1]; q2 = gsrc[2]; q3 = gsrc[3];
    dst[8]  = make_float2(q0.x, q0.y); dst[9]  = make_float2(q0.z, q0.w);
    dst[10] = make_float2(q1.x, q1.y); dst[11] = make_float2(q1.z, q1.w);
    dst[12] = make_float2(q2.x, q2.y); dst[13] = make_float2(q2.z, q2.w);
    dst[14] = make_float2(q3.x, q3.y); dst[15] = make_float2(q3.z, q3.w);
    __syncthreads();
#endif

    v8f acc0 = {}, acc1 = {};
#pragma unroll
    for (int kk = 0; kk < KSTEPS; ++kk) {  // fully unrolled: areg stays in VGPRs
      v2f b0 = bp0[2 * kk];
      v2f b1 = bp1[2 * kk];
      acc0 = __builtin_amdgcn_wmma_f32_16x16x4_f32(false, areg[kk], false, b0,
                                                   (short)0, acc0, false, false);
      acc1 = __builtin_amdgcn_wmma_f32_16x16x4_f32(false, areg[kk], false, b1,
                                                   (short)0, acc1, false, false);
    }

    // Epilogue: two columns per lane
    float sqj0 = sq[j0 + n];
    int   tj0  = tgt[j0 + n];
    float sqj1 = sq[j0 + 16 + n];
    int   tj1  = tgt[j0 + 16 + n];
#pragma unroll
    for (int v = 0; v < 8; ++v) {
      float d0 = fmaxf(fmaf(-2.0f, acc0[v], sqi[v] + sqj0), EPSF);
      if (ti[v] == tj0) pos[v] = fmaxf(pos[v], d0);
      else              neg[v] = fminf(neg[v], d0);
      float d1 = fmaxf(fmaf(-2.0f, acc1[v], sqi[v] + sqj1), EPSF);
      if (ti[v] == tj1) pos[v] = fmaxf(pos[v], d1);
      else              neg[v] = fminf(neg[v], d1);
    }
  }

  // Reduce over the 16 lanes of each half-wave (same row set per half)
#pragma unroll
  for (int v = 0; v < 8; ++v) {
#pragma unroll
    for (int m = 8; m >= 1; m >>= 1) {
      pos[v] = fmaxf(pos[v], __shfl_xor(pos[v], m, 32));
      neg[v] = fminf(neg[v], __shfl_xor(neg[v], m, 32));
    }
  }
  if (n == 0) {  // lane 0 and lane 16 publish their 8 rows each
#pragma unroll
    for (int v = 0; v < 8; ++v) {
      int r = i0 + v + 8 * khalf;
      atomicMax(posA + r, __float_as_int(pos[v]));
      atomicMin(negA + r, __float_as_int(neg[v]));
    }
  }
}

// ---------------------------------------------------------------- final loss
__global__ __launch_bounds__(256) void tl_finish(const int* __restrict__ posA,
                                                 const int* __restrict__ negA,
                                                 float* __restrict__ out) {
  int r   = blockIdx.x * 256 + threadIdx.x;
  float p = __int_as_float(posA[r]);
  float q = __int_as_float(negA[r]);
  float v = fmaxf(p - q + MARGINF, 0.0f);
#pragma unroll
  for (int m = 16; m >= 1; m >>= 1) v += __shfl_xor(v, m, 32);
  __shared__ float part[8];
  if ((threadIdx.x & 31) == 0) part[threadIdx.x >> 5] = v;
  __syncthreads();
  if (threadIdx.x == 0) {
    float s = 0.0f;
#pragma unroll
    for (int w = 0; w < 8; ++w) s += part[w];
    atomicAdd(out, s * (1.0f / (float)BATCH));
  }
}

// ---------------------------------------------------------------- launch
extern "C" void kernel_launch(void* const* d_in, const int* in_sizes, int n_in,
                              void* d_out, int out_size, void* d_ws, size_t ws_size,
                              hipStream_t stream) {
  const float* x   = (const float*)d_in[0];  // [8192, 256] fp32
  const int*   tgt = (const int*)d_in[1];    // [8192] class ids
  float*       out = (float*)d_out;          // scalar

  // Workspace layout: vs (8 MB) | sq | posA | negA  (~8.1 MB total)
  float* vs   = (float*)d_ws;
  float* sqv  = vs + (size_t)BATCH * DIM;
  int*   posA = (int*)(sqv + BATCH);
  int*   negA = posA + BATCH;

  tl_init<<<BATCH / 256, 256, 0, stream>>>(posA, negA, out);
  tl_normalize<<<BATCH / 8, 256, 0, stream>>>(x, vs, sqv);
  dim3 grid(BATCH / BLK_M, BATCH / JCHUNK);  // (64, 16) = 1024 blocks
  tl_mine<<<grid, 256, 0, stream>>>(vs, sqv, tgt, posA, negA);
  tl_finish<<<BATCH / 256, 256, 0, stream>>>(posA, negA, out);
}